// EnhanceDiversityFeatureExtracition_56152402427930
// MI455X (gfx1250) — compile-verified
//
#include <hip/hip_runtime.h>
#include <stdint.h>

// ---------------------------------------------------------------------------
// Problem constants (from reference):
//   x,y: 2048*4096 fp32 ; conv_w: [512,512,64,3] fp32
//   v[f][c] = conv_w[o][i][f][kw],  c = (o*512+i)*3 + kw
//   flat addr(f,c) = (c/3)*192 + f*3 + (c%3)
// ---------------------------------------------------------------------------
#define NXY        8388608          // 2048*4096
#define F_DIM      64
#define QTOT       262144           // 512*512
#define CHUNK_Q    32               // q-groups per LDS chunk
#define CHUNK_K    96               // CHUNK_Q*3, multiple of 4 for WMMA
#define NCHUNKS    (QTOT / CHUNK_Q) // 8192
#define GRAM_BLOCKS 256
#define CHUNKS_PER_BLOCK (NCHUNKS / GRAM_BLOCKS) // 32
#define LDS_STRIDE 100              // 96 + 4 pad (floats), keeps 8B alignment

typedef float v2f __attribute__((ext_vector_type(2)));
typedef float v8f __attribute__((ext_vector_type(8)));

// Workspace layout (floats): ws[0] = MSE sum ; ws[16..16+4095] = gram 64x64
#define WS_GRAM_OFF 16
#define WS_FLOATS   (WS_GRAM_OFF + F_DIM * F_DIM)

#if __has_builtin(__builtin_amdgcn_global_load_async_to_lds_b32)
#define HAS_ASYNC_LDS 1
#else
#define HAS_ASYNC_LDS 0
#endif

typedef __attribute__((address_space(1))) int* gas1_ip;   // global (AS1) int*
typedef __attribute__((address_space(3))) int* as3_ip;    // LDS (AS3) int*

__device__ __forceinline__ void wait_async0() {
#if HAS_ASYNC_LDS
#if __has_builtin(__builtin_amdgcn_s_wait_asynccnt)
    __builtin_amdgcn_s_wait_asynccnt(0);
#else
    asm volatile("s_wait_asynccnt 0x0" ::: "memory");
#endif
#endif
}

__global__ void __launch_bounds__(256)
zero_ws_kernel(float* __restrict__ p, int n) {
    int i = blockIdx.x * blockDim.x + threadIdx.x;
    if (i < n) p[i] = 0.0f;
}

// ---------------------------------------------------------------------------
// MSE partial sum: float4 (b128) loads for bandwidth, wave32 shuffle reduce,
// one global_atomic_add_f32 per block.
// ---------------------------------------------------------------------------
__global__ void __launch_bounds__(256)
mse_kernel(const float4* __restrict__ x, const float4* __restrict__ y,
           float* __restrict__ acc, int n4) {
    __shared__ float red[8];
    int idx    = blockIdx.x * blockDim.x + threadIdx.x;
    int stride = gridDim.x * blockDim.x;
    float s = 0.0f;
    for (int i = idx; i < n4; i += stride) {
        float4 a = x[i];
        float4 b = y[i];
        float dx = a.x - b.x, dy = a.y - b.y, dz = a.z - b.z, dw = a.w - b.w;
        s = fmaf(dx, dx, s);
        s = fmaf(dy, dy, s);
        s = fmaf(dz, dz, s);
        s = fmaf(dw, dw, s);
    }
    for (int off = 16; off > 0; off >>= 1) s += __shfl_down(s, off, 32);
    int lane = threadIdx.x & 31;
    int wv   = threadIdx.x >> 5;
    if (lane == 0) red[wv] = s;
    __syncthreads();
    if (wv == 0) {
        float t = (lane < 8) ? red[lane] : 0.0f;
        for (int off = 4; off > 0; off >>= 1) t += __shfl_down(t, off, 32);
        if (lane == 0) unsafeAtomicAdd(acc, t);
    }
}

// ---------------------------------------------------------------------------
// Stage one K-chunk (64 rows x 96 floats) of v into an LDS buffer.
// Async path: global_load_async_to_lds_b32 (ASYNCcnt), imm offset applies to
// both global and LDS address (ISA §4.4), covering each contiguous triplet.
// Fallback: batched global loads into registers, then ds_stores (no per-load
// s_wait_loadcnt serialization).
// ---------------------------------------------------------------------------
__device__ __forceinline__ void stage_chunk(const float* __restrict__ w,
                                            float* __restrict__ buf,
                                            int q0, int tid) {
#if HAS_ASYNC_LDS
    #pragma unroll
    for (int j = 0; j < 4; ++j) {
        int p  = tid + j * 512;      // 0..2047 (q,f) pairs
        int f  = p & 63;
        int qq = p >> 6;
        const float* src = w + (size_t)(q0 + qq) * 192 + f * 3;
        float*       dst = &buf[f * LDS_STRIDE + qq * 3];
        gas1_ip gsrc = (gas1_ip)(uintptr_t)src;
        as3_ip  ldst = (as3_ip)(uint32_t)(uintptr_t)dst;
        __builtin_amdgcn_global_load_async_to_lds_b32(gsrc, ldst, 0, 0);
        __builtin_amdgcn_global_load_async_to_lds_b32(gsrc, ldst, 4, 0);
        __builtin_amdgcn_global_load_async_to_lds_b32(gsrc, ldst, 8, 0);
        __builtin_prefetch(src + CHUNK_Q * 192, 0, 1); // next chunk
    }
#else
    float t0[4], t1[4], t2[4];
    #pragma unroll
    for (int j = 0; j < 4; ++j) {
        int p  = tid + j * 512;
        int f  = p & 63;
        int qq = p >> 6;
        const float* src = w + (size_t)(q0 + qq) * 192 + f * 3;
        t0[j] = src[0];
        t1[j] = src[1];
        t2[j] = src[2];
        __builtin_prefetch(src + CHUNK_Q * 192, 0, 1);
    }
    #pragma unroll
    for (int j = 0; j < 4; ++j) {
        int p  = tid + j * 512;
        int f  = p & 63;
        int qq = p >> 6;
        float* dst = &buf[f * LDS_STRIDE + qq * 3];
        dst[0] = t0[j];
        dst[1] = t1[j];
        dst[2] = t2[j];
    }
#endif
}

// ---------------------------------------------------------------------------
// Gram = v * v^T via V_WMMA_F32_16X16X4_F32, double-buffered LDS.
// Block = 512 threads = 16 waves; wave w owns output tile (mi,nj)=(w>>2,w&3).
// Pipeline per block: stage chunk c+1 (async) | 24 fp32 WMMAs on chunk c |
// s_wait_asynccnt 0 | workgroup barrier. Partial tiles merged via f32 atomics.
//
// Operand layout (ISA 7.12.2, 32-bit A 16x4 / B 4x16):
//   lanes 0-15 : VGPR0=K0, VGPR1=K1 ; lanes 16-31 : VGPR0=K2, VGPR1=K3
//   -> per lane: ds_load_b64 at row*STRIDE + kk + 2*(lane>=16)
// D layout: VGPR r = row (r + 8*(lane>=16)), col = lane&15.
// ---------------------------------------------------------------------------
__global__ void __launch_bounds__(512)
gram_wmma_kernel(const float* __restrict__ w, float* __restrict__ gram) {
    __shared__ float lds[2][F_DIM * LDS_STRIDE];

    const int tid  = threadIdx.x;
    const int lane = tid & 31;
    const int wave = tid >> 5;
    const int mi   = wave >> 2;
    const int nj   = wave & 3;
    const int r16  = lane & 15;
    const int ko   = (lane >> 4) << 1;   // 0 or 2

    const int aoff = (mi * 16 + r16) * LDS_STRIDE + ko;
    const int boff = (nj * 16 + r16) * LDS_STRIDE + ko;

    v8f acc = {};

    const int chunk0 = blockIdx.x * CHUNKS_PER_BLOCK;

    // prologue: stage chunk 0 into buffer 0
    stage_chunk(w, &lds[0][0], chunk0 * CHUNK_Q, tid);
    wait_async0();
    __syncthreads();

    for (int c = 0; c < CHUNKS_PER_BLOCK; ++c) {
        if (c + 1 < CHUNKS_PER_BLOCK) {
            stage_chunk(w, &lds[(c + 1) & 1][0],
                        (chunk0 + c + 1) * CHUNK_Q, tid);
        }
        const float* arow = &lds[c & 1][aoff];
        const float* brow = &lds[c & 1][boff];
        #pragma unroll
        for (int kk = 0; kk < CHUNK_K; kk += 4) {
            v2f a = *(const v2f*)(arow + kk);
            v2f b = *(const v2f*)(brow + kk);
            acc = __builtin_amdgcn_wmma_f32_16x16x4_f32(
                /*neg_a=*/false, a, /*neg_b=*/false, b,
                /*c_mod=*/(short)0, acc, /*reuse_a=*/false, /*reuse_b=*/false);
        }
        wait_async0();
        __syncthreads();
    }

    const int col   = nj * 16 + r16;
    const int rbase = mi * 16 + ((lane >> 4) << 3);
    #pragma unroll
    for (int r = 0; r < 8; ++r) {
        unsafeAtomicAdd(&gram[(rbase + r) * F_DIM + col], acc[r]);
    }
}

// ---------------------------------------------------------------------------
// Finalize: norms from gram diagonal, cosine-sim mask-sum, fuse MSE mean.
// ---------------------------------------------------------------------------
__global__ void __launch_bounds__(64)
finalize_kernel(const float* __restrict__ ws, float* __restrict__ out) {
    __shared__ float norms[F_DIM];
    __shared__ float partial[F_DIM];
    const float* gram = ws + WS_GRAM_OFF;
    const int i = threadIdx.x;

    norms[i] = sqrtf(gram[i * F_DIM + i]);
    __syncthreads();

    const float ni = norms[i];
    float s = 0.0f;
    for (int j = 0; j < F_DIM; ++j) {
        if (j == i) continue;
        float sim = gram[i * F_DIM + j] / (ni * norms[j]);
        if (sim > 0.2f && sim <= 1.0f) s += sim;
    }
    partial[i] = s;
    __syncthreads();
    for (int off = 32; off > 0; off >>= 1) {
        if (i < off) partial[i] += partial[i + off];
        __syncthreads();
    }
    if (i == 0) {
        out[0] = ws[0] * (1.0f / (float)NXY) + 0.0005f * partial[0];
    }
}

// ---------------------------------------------------------------------------
extern "C" void kernel_launch(void* const* d_in, const int* in_sizes, int n_in,
                              void* d_out, int out_size, void* d_ws, size_t ws_size,
                              hipStream_t stream) {
    const float* x = (const float*)d_in[0];
    const float* y = (const float*)d_in[1];
    const float* w = (const float*)d_in[2];
    float* out = (float*)d_out;
    float* ws  = (float*)d_ws;

    zero_ws_kernel<<<(WS_FLOATS + 255) / 256, 256, 0, stream>>>(ws, WS_FLOATS);
    mse_kernel<<<1024, 256, 0, stream>>>((const float4*)x, (const float4*)y,
                                         ws, NXY / 4);
    gram_wmma_kernel<<<GRAM_BLOCKS, 512, 0, stream>>>(w, ws + WS_GRAM_OFF);
    finalize_kernel<<<1, 64, 0, stream>>>(ws, out);
}